// AttentionFlow_70506183131673
// MI455X (gfx1250) — compile-verified
//
#include <hip/hip_runtime.h>
#include <hip/hip_bf16.h>
#include <math.h>

// Problem constants (B*T folded): N=16 sequences, S=128, D=H=256, 3H=768.
#define NSEQ 16
#define SLEN 128
#define DIM  256
#define H3   768

typedef float v2f __attribute__((ext_vector_type(2)));
typedef float v8f __attribute__((ext_vector_type(8)));

// ---------------------------------------------------------------------------
// Wave-level fp32 WMMA helper: one wave computes a 16x16 f32 tile of
// A(16xK, lda) * B(Kx16, ldb), accumulating into acc, K stepped by 4 with
// V_WMMA_F32_16X16X4_F32 (exact fp32 multiply-accumulate).
//
// Fragment layouts per CDNA5 ISA 7.12.2 (wave32):
//   A 16x4:  lane L holds row m=L&15; VGPR0/1 = K = (L>>4)*2 + {0,1}
//   B 4x16:  lane L holds col n=L&15; VGPR0/1 = K = (L>>4)*2 + {0,1}
//   C 16x16: lane L, VGPR j -> row (j + (L>>4)*8), col (L&15)
// ---------------------------------------------------------------------------
__device__ __forceinline__ v8f wave_gemm_f32(const float* __restrict__ A, int lda,
                                             const float* __restrict__ B, int ldb,
                                             int K, v8f acc, int lane) {
  const int m     = lane & 15;
  const int khalf = (lane >> 4) << 1;
  const int nc    = lane & 15;
  for (int k0 = 0; k0 < K; k0 += 4) {
    v2f a, b;
    a.x = A[m * lda + k0 + khalf];
    a.y = A[m * lda + k0 + khalf + 1];
    b.x = B[(k0 + khalf) * ldb + nc];
    b.y = B[(k0 + khalf + 1) * ldb + nc];
    acc = __builtin_amdgcn_wmma_f32_16x16x4_f32(false, a, false, b,
                                                (short)0, acc, false, false);
  }
  return acc;
}

// ---------------------------------------------------------------------------
// Kernel 1: left = x@w1, right = x@w2, xp = x@K_top + b_in
// One 32-thread (1 wave) block per 16x16 output tile.
// Tiles per seq: left 8x16=128, right 128, xp 8x48=384 -> 640; x16 seq = 10240.
// ---------------------------------------------------------------------------
__global__ __launch_bounds__(32)
void proj_kernel(const float* __restrict__ x,
                 const float* __restrict__ w1, const float* __restrict__ w2,
                 const float* __restrict__ gk, const float* __restrict__ b_in,
                 float* __restrict__ left, float* __restrict__ right,
                 float* __restrict__ xp) {
  const int tile = blockIdx.x;
  const int seq  = tile / 640;
  const int r    = tile % 640;
  const int lane = threadIdx.x;

  const float* A;
  const float* Bm;
  float*       Dst;
  int ldb, ldd;
  float badd_base = 0.0f;
  bool  addb = false;
  int   nt;

  if (r < 256) {                       // left / right GEMMs (N=256)
    const int which = r >> 7;          // 0 = left, 1 = right
    const int rr    = r & 127;
    const int mt    = rr >> 4;
    nt  = rr & 15;
    A   = x + (size_t)seq * SLEN * DIM + mt * 16 * DIM;
    Bm  = (which ? w2 : w1) + nt * 16;
    ldb = DIM;
    Dst = (which ? right : left) + (size_t)seq * SLEN * DIM + mt * 16 * DIM + nt * 16;
    ldd = DIM;
  } else {                             // xp = x @ K_top + b_in  (N=768)
    const int rr = r - 256;
    const int mt = rr / 48;
    nt  = rr % 48;
    A   = x + (size_t)seq * SLEN * DIM + mt * 16 * DIM;
    Bm  = gk + nt * 16;                // K_top = gru_kernel rows [0,256)
    ldb = H3;
    Dst = xp + (size_t)seq * SLEN * H3 + mt * 16 * H3 + nt * 16;
    ldd = H3;
    addb = true;
  }

  v8f acc = {};
  acc = wave_gemm_f32(A, DIM, Bm, ldb, DIM, acc, lane);

  const int nc    = lane & 15;
  const int rbase = (lane >> 4) * 8;
  if (addb) badd_base = b_in[nt * 16 + nc];
#pragma unroll
  for (int j = 0; j < 8; ++j)
    Dst[(rbase + j) * ldd + nc] = acc[j] + badd_base;
}

// ---------------------------------------------------------------------------
// Kernel 2: Bahdanau attention + context.
// One 128-thread block per (seq, query row m).  Thread t owns key j=t:
//   s_j = sum_d v3[d] * tanh(left[m,d] + bias[d] + right[j,d])
// then LDS softmax over j, then coalesced c[m,:] = sum_j p[j] * x[j,:].
// ---------------------------------------------------------------------------
__global__ __launch_bounds__(128)
void attn_kernel(const float* __restrict__ x,
                 const float* __restrict__ left, const float* __restrict__ right,
                 const float* __restrict__ bias, const float* __restrict__ v3,
                 float* __restrict__ cbuf) {
  __shared__ float lb[DIM];    // left[m,:] + bias
  __shared__ float v3s[DIM];
  __shared__ float red[SLEN];
  __shared__ float p[SLEN];

  const int n    = blockIdx.x >> 7;
  const int mrow = blockIdx.x & 127;
  const int t    = threadIdx.x;

  const float* Lrow = left + ((size_t)n * SLEN + mrow) * DIM;
  lb[t]        = Lrow[t] + bias[t];
  lb[t + 128]  = Lrow[t + 128] + bias[t + 128];
  v3s[t]       = v3[t];
  v3s[t + 128] = v3[t + 128];
  __syncthreads();

  // --- scores ---
  const float* R = right + ((size_t)n * SLEN + t) * DIM;
  float s = 0.0f;
  for (int d = 0; d < DIM; d += 4) {
    float4 rv = *(const float4*)(R + d);
    s += v3s[d]     * tanhf(lb[d]     + rv.x);
    s += v3s[d + 1] * tanhf(lb[d + 1] + rv.y);
    s += v3s[d + 2] * tanhf(lb[d + 2] + rv.z);
    s += v3s[d + 3] * tanhf(lb[d + 3] + rv.w);
  }

  // --- softmax over j (numerically stable) ---
  red[t] = s;
  __syncthreads();
  for (int off = 64; off > 0; off >>= 1) {
    if (t < off) red[t] = fmaxf(red[t], red[t + off]);
    __syncthreads();
  }
  const float mx = red[0];
  __syncthreads();
  const float e = expf(s - mx);
  red[t] = e;
  __syncthreads();
  for (int off = 64; off > 0; off >>= 1) {
    if (t < off) red[t] += red[t + off];
    __syncthreads();
  }
  p[t] = e / red[0];
  __syncthreads();

  // --- context: c[m,dd] = sum_j p[j] * x[j,dd], coalesced over dd ---
  const float* Xn = x + (size_t)n * SLEN * DIM;
  for (int dd = t; dd < DIM; dd += 128) {
    float acc = 0.0f;
    for (int j = 0; j < SLEN; ++j) acc = fmaf(p[j], Xn[j * DIM + dd], acc);
    cbuf[((size_t)n * SLEN + mrow) * DIM + dd] = acc;
  }
}

// ---------------------------------------------------------------------------
// Kernel 3: xp += c @ K_bot   (K_bot = gru_kernel rows [256,512))
// One wave per 16x16 tile; accumulator seeded from xp. 16*8*48 = 6144 tiles.
// ---------------------------------------------------------------------------
__global__ __launch_bounds__(32)
void ctx_gemm_kernel(const float* __restrict__ cbuf,
                     const float* __restrict__ gk,
                     float* __restrict__ xp) {
  const int tile = blockIdx.x;
  const int seq  = tile / 384;
  const int rr   = tile % 384;
  const int mt   = rr / 48;
  const int nt   = rr % 48;
  const int lane = threadIdx.x;

  const float* A  = cbuf + (size_t)seq * SLEN * DIM + mt * 16 * DIM;
  const float* Bm = gk + (size_t)DIM * H3 + nt * 16;   // bottom half of kernel
  float*       P  = xp + (size_t)seq * SLEN * H3 + mt * 16 * H3 + nt * 16;

  const int nc    = lane & 15;
  const int rbase = (lane >> 4) * 8;

  v8f acc = {};
#pragma unroll
  for (int j = 0; j < 8; ++j) acc[j] = P[(rbase + j) * H3 + nc];

  acc = wave_gemm_f32(A, DIM, Bm, H3, DIM, acc, lane);

#pragma unroll
  for (int j = 0; j < 8; ++j) P[(rbase + j) * H3 + nc] = acc[j];
}

// ---------------------------------------------------------------------------
// Kernel 4: GRU scan, all 16 sequences batched so M=16 is one WMMA tile row.
// Single block, 512 threads = 16 waves; h[16x256] and rp[16x768] live in LDS
// (64 KB). Per step: 16 waves x 3 n-tiles of rp = h @ rec_kernel + b_rec via
// fp32 WMMA, barrier, elementwise gates, h update + output store, barrier.
// ---------------------------------------------------------------------------
__global__ __launch_bounds__(512)
void gru_scan_kernel(const float* __restrict__ xp,
                     const float* __restrict__ rec,
                     const float* __restrict__ b_rec,
                     float* __restrict__ out) {
  __shared__ float hbuf[NSEQ * DIM];   // 16 KB
  __shared__ float rp[NSEQ * H3];      // 48 KB

  const int t = threadIdx.x;
  for (int i = t; i < NSEQ * DIM; i += 512) hbuf[i] = 0.0f;   // h0 = 0
  __syncthreads();

  const int wave  = t >> 5;
  const int lane  = t & 31;
  const int m     = lane & 15;         // sequence index (A row)
  const int khalf = (lane >> 4) << 1;
  const int nc    = lane & 15;
  const int rbase = (lane >> 4) * 8;

  for (int step = 0; step < SLEN; ++step) {
    // ---- rp = h @ rec_kernel + b_rec : 48 n-tiles over 16 waves ----
#pragma unroll
    for (int tt = 0; tt < 3; ++tt) {
      const int nt = wave + tt * 16;
      const float* Bm = rec + nt * 16;
      v8f acc = {};
      for (int k0 = 0; k0 < DIM; k0 += 4) {
        v2f a, b;
        a.x = hbuf[m * DIM + k0 + khalf];
        a.y = hbuf[m * DIM + k0 + khalf + 1];
        b.x = Bm[(k0 + khalf) * H3 + nc];
        b.y = Bm[(k0 + khalf + 1) * H3 + nc];
        acc = __builtin_amdgcn_wmma_f32_16x16x4_f32(false, a, false, b,
                                                    (short)0, acc, false, false);
      }
      const float bb = b_rec[nt * 16 + nc];
#pragma unroll
      for (int j = 0; j < 8; ++j)
        rp[(rbase + j) * H3 + nt * 16 + nc] = acc[j] + bb;
    }
    __syncthreads();

    // ---- gates: 16*256 hidden elements over 512 threads ----
    for (int e = t; e < NSEQ * DIM; e += 512) {
      const int sq = e >> 8;
      const int d  = e & 255;
      const float* xrow = xp + ((size_t)sq * SLEN + step) * H3;
      const float xz = xrow[d];
      const float xr = xrow[DIM + d];
      const float xh = xrow[2 * DIM + d];
      const float rz = rp[sq * H3 + d];
      const float rr = rp[sq * H3 + DIM + d];
      const float rh = rp[sq * H3 + 2 * DIM + d];
      const float z  = 1.0f / (1.0f + expf(-(xz + rz)));
      const float r  = 1.0f / (1.0f + expf(-(xr + rr)));
      const float hh = tanhf(fmaf(r, rh, xh));
      const float hn = z * hbuf[e] + (1.0f - z) * hh;   // reads own h only
      out[((size_t)sq * SLEN + step) * DIM + d] = hn;
      hbuf[e] = hn;
    }
    __syncthreads();
  }
}

// ---------------------------------------------------------------------------
// Host launcher
// ---------------------------------------------------------------------------
extern "C" void kernel_launch(void* const* d_in, const int* in_sizes, int n_in,
                              void* d_out, int out_size, void* d_ws, size_t ws_size,
                              hipStream_t stream) {
  (void)in_sizes; (void)n_in; (void)out_size; (void)ws_size;

  const float* features = (const float*)d_in[0];   // [16,128,256]
  const float* w1       = (const float*)d_in[1];   // [256,256]
  const float* w2       = (const float*)d_in[2];   // [256,256]
  const float* bias     = (const float*)d_in[3];   // [256]
  const float* v3       = (const float*)d_in[4];   // [256]
  const float* gk       = (const float*)d_in[5];   // [512,768]
  const float* grk      = (const float*)d_in[6];   // [256,768]
  const float* gb       = (const float*)d_in[7];   // [2,768]
  float*       out      = (float*)d_out;           // [16,128,256]

  float* ws    = (float*)d_ws;
  float* left  = ws;                                   // 16*128*256
  float* right = left  + (size_t)NSEQ * SLEN * DIM;    // 16*128*256
  float* cbuf  = right + (size_t)NSEQ * SLEN * DIM;    // 16*128*256
  float* xp    = cbuf  + (size_t)NSEQ * SLEN * DIM;    // 16*128*768

  proj_kernel<<<NSEQ * 640, 32, 0, stream>>>(features, w1, w2, gk, gb,
                                             left, right, xp);
  attn_kernel<<<NSEQ * SLEN, 128, 0, stream>>>(features, left, right,
                                               bias, v3, cbuf);
  ctx_gemm_kernel<<<NSEQ * 384, 32, 0, stream>>>(cbuf, gk, xp);
  gru_scan_kernel<<<1, 512, 0, stream>>>(xp, grk, gb + H3, out);
}